// VectorQuantizer_66082366816964
// MI455X (gfx1250) — compile-verified
//
#include <hip/hip_runtime.h>
#include <hip/hip_bf16.h>

typedef __attribute__((ext_vector_type(2))) float v2f;
typedef __attribute__((ext_vector_type(8))) float v8f;

#define DIM            64
#define NCODE          1024
#define NTILES         (NCODE / 16)            // 64 n-tiles
#define WAVES          16
#define THREADS        (WAVES * 32)            // 512 threads = 16 wave32
#define ROWS_PER_WAVE  32                      // two 16-row M-tiles per wave
#define ROWS_PER_BLOCK (WAVES * ROWS_PER_WAVE) // 512 z-rows per block
#define K2             (DIM / 2)               // 32 k-pair rows of float2
#define PSTRIDE        1040                    // float2 stride; padding kills LDS bank conflicts
#define P_FLOATS       (K2 * PSTRIDE * 2)      // 66560 floats = 266240 B
#define CSQ_OFF        P_FLOATS
#define IDX_OFF        (P_FLOATS + NCODE)
#define SMEM_FLOATS    (IDX_OFF + WAVES * ROWS_PER_WAVE)
#define SMEM_BYTES     (SMEM_FLOATS * 4)       // 272384 B <= 320 KB/WGP

__global__ __launch_bounds__(THREADS) void vq_wmma_kernel(
    const float* __restrict__ z, const float* __restrict__ codebook,
    float* __restrict__ out_q, float* __restrict__ out_idx)
{
    extern __shared__ float smem[];
    float2* P   = (float2*)smem;                 // [K2][PSTRIDE] transposed codebook pairs
    float* csq  = smem + CSQ_OFF;                // [NCODE]
    int*   idxS = (int*)(smem + IDX_OFF);        // [WAVES][32]

    const int tid = threadIdx.x;

    // ---- Stage codebook into LDS, transposed into (k-pair, n) float2 layout ----
    for (int c = tid; c < NCODE * (DIM / 4); c += THREADS) {
        const int n = c >> 4;                    // codebook row
        const int j = c & 15;                    // float4 chunk: k = 4j..4j+3
        const float4 v = ((const float4*)codebook)[c];
        P[(2 * j + 0) * PSTRIDE + n] = make_float2(v.x, v.y);
        P[(2 * j + 1) * PSTRIDE + n] = make_float2(v.z, v.w);
    }
    __syncthreads();

    // ---- ||c||^2 from LDS ----
    for (int n = tid; n < NCODE; n += THREADS) {
        float s = 0.f;
        #pragma unroll 8
        for (int k2 = 0; k2 < K2; ++k2) {
            const float2 p = P[k2 * PSTRIDE + n];
            s = fmaf(p.x, p.x, fmaf(p.y, p.y, s));
        }
        csq[n] = s;
    }
    __syncthreads();

    const int wave = tid >> 5;
    const int lane = tid & 31;
    const int m    = lane & 15;                  // row within 16-row tile / col within n-tile
    const int kh   = lane >> 4;                  // lane-half selects K pair within K=4 slice

    const int rowBase = blockIdx.x * ROWS_PER_BLOCK + wave * ROWS_PER_WAVE;

    // ---- A fragments (16x4 f32 layout) for two M-tiles, scaled by -2 ----
    v2f afrag0[16], afrag1[16];
    const float* zrow0 = z + (size_t)(rowBase + m) * DIM;
    const float* zrow1 = z + (size_t)(rowBase + 16 + m) * DIM;
    #pragma unroll
    for (int j = 0; j < 16; ++j) {
        const float2 t0 = *(const float2*)(zrow0 + 4 * j + 2 * kh);
        const float2 t1 = *(const float2*)(zrow1 + 4 * j + 2 * kh);
        afrag0[j].x = -2.0f * t0.x;  afrag0[j].y = -2.0f * t0.y;
        afrag1[j].x = -2.0f * t1.x;  afrag1[j].y = -2.0f * t1.y;
    }

    float bestV0[8], bestV1[8];
    int   bestI0[8], bestI1[8];
    #pragma unroll
    for (int r = 0; r < 8; ++r) {
        bestV0[r] = 3.4e38f; bestI0[r] = 0;
        bestV1[r] = 3.4e38f; bestI1[r] = 0;
    }

    // ---- 64 n-tiles: stage all 16 B fragments into distinct regs, then
    //      drain them through two interleaved WMMA accumulator chains ----
    for (int t = 0; t < NTILES; ++t) {
        const int n = t * 16 + m;

        v2f bfrag[16];                            // 16 live ds_load_b64 results
        #pragma unroll
        for (int j = 0; j < 16; ++j) {
            const float2 bb = P[(2 * j + kh) * PSTRIDE + n];
            bfrag[j].x = bb.x;  bfrag[j].y = bb.y;
        }
        const float cs = csq[n];

        v8f acc0 = {}, acc1 = {};
        #pragma unroll
        for (int j = 0; j < 16; ++j) {
            acc0 = __builtin_amdgcn_wmma_f32_16x16x4_f32(
                false, afrag0[j], false, bfrag[j], (short)0, acc0, false, false);
            acc1 = __builtin_amdgcn_wmma_f32_16x16x4_f32(
                false, afrag1[j], false, bfrag[j], (short)0, acc1, false, false);
        }

        #pragma unroll
        for (int r = 0; r < 8; ++r) {
            const float d0 = acc0[r] + cs;                    // -2 z.c + ||c||^2
            if (d0 < bestV0[r]) { bestV0[r] = d0; bestI0[r] = n; }
            const float d1 = acc1[r] + cs;
            if (d1 < bestV1[r]) { bestV1[r] = d1; bestI1[r] = n; }
        }
    }

    // ---- min+index butterfly across the 16 lanes sharing each output row ----
    #pragma unroll
    for (int off = 1; off < 16; off <<= 1) {
        #pragma unroll
        for (int r = 0; r < 8; ++r) {
            {
                const float ov = __shfl_xor(bestV0[r], off, 32);
                const int   oi = __shfl_xor(bestI0[r], off, 32);
                if (ov < bestV0[r] || (ov == bestV0[r] && oi < bestI0[r])) {
                    bestV0[r] = ov; bestI0[r] = oi;
                }
            }
            {
                const float ov = __shfl_xor(bestV1[r], off, 32);
                const int   oi = __shfl_xor(bestI1[r], off, 32);
                if (ov < bestV1[r] || (ov == bestV1[r] && oi < bestI1[r])) {
                    bestV1[r] = ov; bestI1[r] = oi;
                }
            }
        }
    }
    if (m == 0) {
        #pragma unroll
        for (int r = 0; r < 8; ++r) {
            idxS[wave * 32 + kh * 8 + r]      = bestI0[r];
            idxS[wave * 32 + 16 + kh * 8 + r] = bestI1[r];
        }
    }
    __syncthreads();

    // ---- outputs: indices (as float) + gather z_q rows from L2-hot codebook ----
    #pragma unroll
    for (int half = 0; half < 2; ++half) {
        const int row = half * 16 + m;                       // 0..31 within wave
        const int idx = idxS[wave * 32 + row];
        if (kh == 0) out_idx[rowBase + row] = (float)idx;
        const float4* src = (const float4*)(codebook + (size_t)idx * DIM + kh * 32);
        float4*       dst = (float4*)(out_q + (size_t)(rowBase + row) * DIM + kh * 32);
        #pragma unroll
        for (int q = 0; q < 8; ++q) dst[q] = src[q];
    }
}

extern "C" void kernel_launch(void* const* d_in, const int* in_sizes, int n_in,
                              void* d_out, int out_size, void* d_ws, size_t ws_size,
                              hipStream_t stream) {
    const float* z  = (const float*)d_in[0];
    const float* cb = (const float*)d_in[1];
    const int B = in_sizes[0] / DIM;             // 262144
    float* out_q   = (float*)d_out;              // B*DIM floats
    float* out_idx = out_q + (size_t)B * DIM;    // B floats (indices as float)

    (void)hipFuncSetAttribute((const void*)vq_wmma_kernel,
                              hipFuncAttributeMaxDynamicSharedMemorySize, SMEM_BYTES);
    vq_wmma_kernel<<<B / ROWS_PER_BLOCK, THREADS, SMEM_BYTES, stream>>>(z, cb, out_q, out_idx);
}